// BFGSCameraSolver_51384988729815
// MI455X (gfx1250) — compile-verified
//
#include <hip/hip_runtime.h>
#include <hip/hip_bf16.h>

typedef __attribute__((ext_vector_type(2))) float v2f;
typedef __attribute__((ext_vector_type(8))) float v8f;

#define NP 128         // parameters per problem
#define MAX_ITERS 5
#define LS_C1 1e-4f
#define LS_C2 0.9f
#define WIDEN_ITERS 4
#define ZOOM_ITERS 8

// full wave32 sum (result broadcast to all lanes)
__device__ __forceinline__ float wave_sum(float v) {
#pragma unroll
    for (int off = 16; off; off >>= 1) v += __shfl_xor(v, off, 32);
    return v;
}
// sum within each 16-lane half of a wave32 (xor bits < 16 stay in the half)
__device__ __forceinline__ float half_sum16(float v) {
#pragma unroll
    for (int off = 8; off; off >>= 1) v += __shfl_xor(v, off, 32);
    return v;
}

__global__ __launch_bounds__(256)
void BFGS_dense_wmma_kernel(const float* __restrict__ x0,
                            const float* __restrict__ qd,
                            const float* __restrict__ qc,
                            const float* __restrict__ H0,
                            float* __restrict__ out) {
    __shared__ float xv[NP], dv[NP], cv[NP];
    __shared__ float g_s[NP], p_s[NP], s_s[NP], y_s[NP], w_s[NP];

    const int tid  = threadIdx.x;
    const int lane = tid & 31;
    const int wid  = tid >> 5;      // wave 0..7 owns H rows [16*wid, 16*wid+16)
    const int hi   = lane >> 4;     // half of wave (WMMA layout: M += 8*hi)
    const int col  = lane & 15;     // N index within tile / column within slice
    const long base = (long)blockIdx.x * NP;

    if (tid < NP) {
        xv[tid] = x0[base + tid];
        dv[tid] = qd[base + tid];
        cv[tid] = qc[base + tid];
    }

    // H register tiles in WMMA f32 16x16 C/D layout:
    // Ht[j][v] = H[16*wid + v + 8*hi][16*j + col]
    v8f Ht[8];
#pragma unroll
    for (int j = 0; j < 8; ++j)
#pragma unroll
        for (int v = 0; v < 8; ++v)
            Ht[j][v] = H0[(16 * wid + v + 8 * hi) * NP + 16 * j + col];
    __syncthreads();

    for (int it = 0; it < MAX_ITERS; ++it) {
        // g = d * (x - c)
        if (tid < NP) g_s[tid] = dv[tid] * (xv[tid] - cv[tid]);
        __syncthreads();

        // p = -H g : each wave's 16-row slice, VALU dots out of C-layout tiles
        {
            float acc[8];
#pragma unroll
            for (int v = 0; v < 8; ++v) acc[v] = 0.f;
#pragma unroll
            for (int j = 0; j < 8; ++j) {
                float gj = g_s[16 * j + col];
#pragma unroll
                for (int v = 0; v < 8; ++v) acc[v] += Ht[j][v] * gj;
            }
#pragma unroll
            for (int v = 0; v < 8; ++v) acc[v] = half_sum16(acc[v]);
            if (col == 0) {
#pragma unroll
                for (int v = 0; v < 8; ++v)
                    p_s[16 * wid + v + 8 * hi] = -acc[v];
            }
        }
        __syncthreads();

        // ---- strong-Wolfe line search: redundantly per wave, 4 elems/lane ----
        float xr[4], dr[4], cr[4], pr[4];
#pragma unroll
        for (int k = 0; k < 4; ++k) {
            int i = lane + 32 * k;
            xr[k] = xv[i]; dr[k] = dv[i]; cr[k] = cv[i]; pr[k] = p_s[i];
        }
        auto evalfg = [&](float a, float& f, float& gd) {
            float fl = 0.f, gl = 0.f;
#pragma unroll
            for (int k = 0; k < 4; ++k) {
                float r = xr[k] + a * pr[k] - cr[k];
                float t = dr[k] * r;
                fl += t * r;
                gl += t * pr[k];
            }
            f  = 0.5f * wave_sum(fl);
            gd = wave_sum(gl);
        };

        float f0, g0;
        evalfg(0.f, f0, g0);
        float a_lo = 0.f, a_hi = 1.f, f_lo_w = f0;
        float zlo = 0.f, zhi = 1.f, f_zlo = f0;
        bool  done = false, zoom = false;
        float a_star = 1.f;
#pragma unroll
        for (int i = 0; i < WIDEN_ITERS; ++i) {
            float fa, ga;
            evalfg(a_hi, fa, ga);
            bool active = !done && !zoom;
            bool armijo_fail = fa > f0 + LS_C1 * a_hi * g0;
            if (i > 0) armijo_fail = armijo_fail || (fa >= f_lo_w);
            bool newly_zoom = active && armijo_fail;
            bool curv_ok = fabsf(ga) <= -LS_C2 * g0;
            bool newly_done = active && !armijo_fail && curv_ok;
            bool pos   = active && !armijo_fail && !curv_ok && (ga >= 0.f);
            bool still = active && !armijo_fail && !curv_ok && (ga < 0.f);
            if (newly_done) { a_star = a_hi; done = true; }
            if (newly_zoom) { zlo = a_lo; zhi = a_hi; f_zlo = f_lo_w; }
            else if (pos)   { zlo = a_hi; zhi = a_lo; f_zlo = fa; }
            zoom = zoom || newly_zoom || pos;
            if (still) { a_lo = a_hi; f_lo_w = fa; a_hi *= 2.f; }
        }
#pragma unroll
        for (int i = 0; i < ZOOM_ITERS; ++i) {
            float aj = 0.5f * (zlo + zhi);
            float fj, gj;
            evalfg(aj, fj, gj);
            bool active = zoom && !done;
            bool hi_fail = (fj > f0 + LS_C1 * aj * g0) || (fj >= f_zlo);
            bool set_hi = active && hi_fail;
            bool curv_ok = fabsf(gj) <= -LS_C2 * g0;
            bool newly_done = active && !hi_fail && curv_ok;
            bool move = active && !hi_fail && !curv_ok;
            bool flip = move && (gj * (zhi - zlo) >= 0.f);
            if (newly_done) { a_star = aj; done = true; }
            float zhi_n = set_hi ? aj : (flip ? zlo : zhi);
            if (move) { zlo = aj; f_zlo = fj; }
            zhi = zhi_n;
        }
        float alpha = done ? a_star : (zoom ? 0.5f * (zlo + zhi) : a_hi);

        // s = alpha*p ; x += s ; y = g_new - g
        if (tid < NP) {
            float sv = alpha * p_s[tid];
            float xn = xv[tid] + sv;
            float gn = dv[tid] * (xn - cv[tid]);
            s_s[tid] = sv;
            y_s[tid] = gn - g_s[tid];
            xv[tid]  = xn;
        }
        __syncthreads();

        // sy (redundant per wave)
        float accs = 0.f;
#pragma unroll
        for (int k = 0; k < 4; ++k) { int i = lane + 32 * k; accs += s_s[i] * y_s[i]; }
        float sy = wave_sum(accs);

        // w = H y : per-wave slice matvec
        {
            float acc[8];
#pragma unroll
            for (int v = 0; v < 8; ++v) acc[v] = 0.f;
#pragma unroll
            for (int j = 0; j < 8; ++j) {
                float yj = y_s[16 * j + col];
#pragma unroll
                for (int v = 0; v < 8; ++v) acc[v] += Ht[j][v] * yj;
            }
#pragma unroll
            for (int v = 0; v < 8; ++v) acc[v] = half_sum16(acc[v]);
            if (col == 0) {
#pragma unroll
                for (int v = 0; v < 8; ++v)
                    w_s[16 * wid + v + 8 * hi] = acc[v];
            }
        }
        __syncthreads();

        float accy = 0.f;
#pragma unroll
        for (int k = 0; k < 4; ++k) { int i = lane + 32 * k; accy += y_s[i] * w_s[i]; }
        float yHy = wave_sum(accy);

        float alpha_c = 0.f, beta_c = 0.f;   // sy==0 -> dH = 0 (WMMA adds zero)
        if (sy != 0.f) { beta_c = 1.f / sy; alpha_c = (sy + yHy) * beta_c * beta_c; }

        // rank-2 BFGS update via WMMA f32 16x16x4:
        // dH = A * B, A = [s_slice, w_slice, 0, 0] (16x4),
        //             B = [a*s^T - b*w^T ; -b*s^T ; 0 ; 0] (4x16)
        // A layout: lanes 0-15 hold K=0(.x)/K=1(.y); lanes 16-31 hold K=2/3 (zero)
        v2f A;
        {
            float sm = s_s[16 * wid + col];
            float wm = w_s[16 * wid + col];
            A.x = hi ? 0.f : sm;
            A.y = hi ? 0.f : wm;
        }
#pragma unroll
        for (int j = 0; j < 8; ++j) {
            float sj = s_s[16 * j + col];
            float wj = w_s[16 * j + col];
            v2f Bv;   // B layout mirrors A: lanes 0-15 rows K=0/K=1, lanes 16-31 zero
            Bv.x = hi ? 0.f : (alpha_c * sj - beta_c * wj);
            Bv.y = hi ? 0.f : (-beta_c * sj);
            Ht[j] = __builtin_amdgcn_wmma_f32_16x16x4_f32(
                false, A, false, Bv, (short)0, Ht[j], false, false);
        }
        __syncthreads();
    }

    if (tid < NP) out[base + tid] = xv[tid];
}

extern "C" void kernel_launch(void* const* d_in, const int* in_sizes, int n_in,
                              void* d_out, int out_size, void* d_ws, size_t ws_size,
                              hipStream_t stream) {
    const float* x0 = (const float*)d_in[0];   // [B,E,P] f32
    const float* qd = (const float*)d_in[1];   // [B,E,P] f32
    const float* qc = (const float*)d_in[2];   // [B,E,P] f32
    const float* H0 = (const float*)d_in[3];   // [P,P]  f32 (shared initial inv-Hessian)
    float* out = (float*)d_out;

    const int nprob = in_sizes[0] / NP;        // B*E = 2048
    BFGS_dense_wmma_kernel<<<dim3(nprob), dim3(256), 0, stream>>>(x0, qd, qc, H0, out);
}